// DecoderLSTM_19181323944577
// MI455X (gfx1250) — compile-verified
//
#include <hip/hip_runtime.h>
#include <hip/hip_bf16.h>
#include <math.h>

// Dims (fixed by reference): B=2048, T=64, m=p=128, L=64, gates=4p=512, cat=384
#define BB   2048
#define TT   64
#define MM   128
#define PP   128
#define GG   512
#define LL   64
#define CAT  384
#define BT   (BB * TT)          // 131072

typedef __attribute__((ext_vector_type(2))) float v2f;
typedef __attribute__((ext_vector_type(8))) float v8f;

// ---------------------------------------------------------------------------
// WMMA f32 16x16x4 fragment helpers (wave32; EXEC must be all-1s -> full
// 256-thread blocks, uniform control flow around the wmma calls).
// A (16x4): lane l -> row l%16, K = 2*(l/16)+{0,1}  (float2 per lane)
// B (4x16): lane l -> col l%16, K = 2*(l/16)+{0,1}
// C/D (16x16): elem r of v8f in lane l -> row r + 8*(l/16), col l%16
// ---------------------------------------------------------------------------
__device__ __forceinline__ v8f wmma_f32(v2f a, v2f b, v8f c) {
    return __builtin_amdgcn_wmma_f32_16x16x4_f32(false, a, false, b,
                                                 (short)0, c, false, false);
}

// ---------------------------------------------------------------------------
// Zero-init scratch (d0 = s0 = 0; ws is poisoned by harness)
// ---------------------------------------------------------------------------
__global__ void zero_kernel(float* __restrict__ p, int n) {
    int i = blockIdx.x * 256 + threadIdx.x;
    if (i < n) p[i] = 0.0f;
}

// ---------------------------------------------------------------------------
// HW[b,t,n] = sum_k H[b,t,k] * FC1_w[n, 256+k] + FC1_b[n]     (done once)
// M = BT = 131072 (16 rows/block), N = 128 (8 waves x 16), K = 128
// ---------------------------------------------------------------------------
__global__ __launch_bounds__(256) void hw_gemm_kernel(
    const float* __restrict__ H, const float* __restrict__ FC1_w,
    const float* __restrict__ FC1_b, float* __restrict__ HW)
{
    const int lane = threadIdx.x & 31, wave = threadIdx.x >> 5;
    const int row0 = blockIdx.x * 16, col0 = wave * 16;
    const int mI = lane & 15, kH = (lane >> 4) * 2;

    const float* arow = H + (long)(row0 + mI) * MM + kH;
    const float* brow = FC1_w + (long)(col0 + mI) * CAT + 256 + kH;

    v8f c = {};
    for (int k0 = 0; k0 < MM; k0 += 4) {
        v2f a; a.x = arow[k0]; a.y = arow[k0 + 1];
        v2f b; b.x = brow[k0]; b.y = brow[k0 + 1];
        c = wmma_f32(a, b, c);
    }
    const int moff = (lane >> 4) * 8;
    const float bias = FC1_b[col0 + mI];
    float* drow = HW + (long)(row0 + moff) * MM + col0 + mI;
#pragma unroll
    for (int r = 0; r < 8; ++r) drow[(long)r * MM] = c[r] + bias;
}

// ---------------------------------------------------------------------------
// v[b,n] = d @ W_dT + s @ W_sT  (W_d = FC1_w[:,0:128], W_s = FC1_w[:,128:256])
// M = 2048 (16/block), N = 128, K = 128+128
// ---------------------------------------------------------------------------
__global__ __launch_bounds__(256) void proj_kernel(
    const float* __restrict__ dm, const float* __restrict__ sm,
    const float* __restrict__ FC1_w, float* __restrict__ vv)
{
    const int lane = threadIdx.x & 31, wave = threadIdx.x >> 5;
    const int row0 = blockIdx.x * 16, col0 = wave * 16;
    const int mI = lane & 15, kH = (lane >> 4) * 2;

    const float* ad = dm + (long)(row0 + mI) * PP + kH;
    const float* as = sm + (long)(row0 + mI) * PP + kH;
    const float* bw = FC1_w + (long)(col0 + mI) * CAT + kH;

    v8f c = {};
    for (int k0 = 0; k0 < PP; k0 += 4) {
        v2f a; a.x = ad[k0]; a.y = ad[k0 + 1];
        v2f b; b.x = bw[k0]; b.y = bw[k0 + 1];
        c = wmma_f32(a, b, c);
    }
    for (int k0 = 0; k0 < PP; k0 += 4) {
        v2f a; a.x = as[k0]; a.y = as[k0 + 1];
        v2f b; b.x = bw[PP + k0]; b.y = bw[PP + k0 + 1];
        c = wmma_f32(a, b, c);
    }
    const int moff = (lane >> 4) * 8;
    float* drow = vv + (long)(row0 + moff) * MM + col0 + mI;
#pragma unroll
    for (int r = 0; r < 8; ++r) drow[(long)r * MM] = c[r];
}

// ---------------------------------------------------------------------------
// Attention (per batch row b): logits -> softmax over T -> C -> lstm_in
// One block of 256 threads (8 waves) per b. HW/H stream from L2 (134MB fits).
// ---------------------------------------------------------------------------
__global__ __launch_bounds__(256) void attn_kernel(
    const float* __restrict__ H, const float* __restrict__ HW,
    const float* __restrict__ vv, const float* __restrict__ FC2_w,
    const float* __restrict__ FC2_b, const float* __restrict__ FCin_w,
    const float* __restrict__ FCin_b, const float* __restrict__ y,
    float* __restrict__ C, float* __restrict__ lstm_in, int t)
{
    __shared__ float sv[MM], sw[MM], slog[TT], salpha[TT], scbuf[MM], sstat[2];
    const int tid = threadIdx.x, b = blockIdx.x;
    const int wave = tid >> 5, lane = tid & 31;

    if (tid < MM) { sv[tid] = vv[b * MM + tid]; sw[tid] = FC2_w[tid]; }
    __syncthreads();

    // logits[t'] = FC2_w . tanh(HW[b,t',:] + v[b,:]) + FC2_b
    const float* hwB = HW + (long)b * (TT * MM);
    for (int tt = wave; tt < TT; tt += 8) {
        const float4 hv = *(const float4*)(hwB + tt * MM + lane * 4);
        float acc = sw[lane * 4 + 0] * tanhf(hv.x + sv[lane * 4 + 0])
                  + sw[lane * 4 + 1] * tanhf(hv.y + sv[lane * 4 + 1])
                  + sw[lane * 4 + 2] * tanhf(hv.z + sv[lane * 4 + 2])
                  + sw[lane * 4 + 3] * tanhf(hv.w + sv[lane * 4 + 3]);
        for (int off = 16; off > 0; off >>= 1) acc += __shfl_xor(acc, off, 32);
        if (lane == 0) slog[tt] = acc + FC2_b[0];
    }
    __syncthreads();

    // softmax over T=64 (single wave reduces)
    if (tid < 32) {
        float a0 = slog[tid], a1 = slog[tid + 32];
        float mx = fmaxf(a0, a1);
        for (int off = 16; off > 0; off >>= 1) mx = fmaxf(mx, __shfl_xor(mx, off, 32));
        float e = __expf(a0 - mx) + __expf(a1 - mx);
        for (int off = 16; off > 0; off >>= 1) e += __shfl_xor(e, off, 32);
        if (tid == 0) { sstat[0] = mx; sstat[1] = e; }
    }
    __syncthreads();
    if (tid < TT) salpha[tid] = __expf(slog[tid] - sstat[0]) / sstat[1];
    __syncthreads();

    // C[b,j] = sum_t alpha[t] * H[b,t,j]   (j coalesced across threads)
    if (tid < MM) {
        const float* hB = H + (long)b * (TT * MM) + tid;
        float acc = 0.0f;
        for (int tt = 0; tt < TT; ++tt) acc += salpha[tt] * hB[tt * MM];
        C[b * MM + tid] = acc;
        scbuf[tid] = acc * FCin_w[1 + tid];
    }
    __syncthreads();

    // lstm_in[b] = FCin_w[0]*y[b,t] + FCin_w[1:] . C + FCin_b
    if (tid < 32) {
        float p = scbuf[tid] + scbuf[tid + 32] + scbuf[tid + 64] + scbuf[tid + 96];
        for (int off = 16; off > 0; off >>= 1) p += __shfl_xor(p, off, 32);
        if (tid == 0)
            lstm_in[b] = p + FCin_w[0] * y[b * LL + t] + FCin_b[0];
    }
}

// ---------------------------------------------------------------------------
// gates[b,n] = d @ W_hhT + lstm_in[b]*W_ih[n] + b_ih[n] + b_hh[n]
// M = 2048 (16/block), N = 512 (grid.y = 4 groups of 128), K = 128
// ---------------------------------------------------------------------------
__global__ __launch_bounds__(256) void gates_gemm_kernel(
    const float* __restrict__ dm, const float* __restrict__ W_hh,
    const float* __restrict__ W_ih, const float* __restrict__ b_ih,
    const float* __restrict__ b_hh, const float* __restrict__ lstm_in,
    float* __restrict__ gates)
{
    const int lane = threadIdx.x & 31, wave = threadIdx.x >> 5;
    const int row0 = blockIdx.x * 16;
    const int col0 = blockIdx.y * 128 + wave * 16;
    const int mI = lane & 15, kH = (lane >> 4) * 2;

    const float* arow = dm + (long)(row0 + mI) * PP + kH;
    const float* brow = W_hh + (long)(col0 + mI) * PP + kH;

    v8f c = {};
    for (int k0 = 0; k0 < PP; k0 += 4) {
        v2f a; a.x = arow[k0]; a.y = arow[k0 + 1];
        v2f b; b.x = brow[k0]; b.y = brow[k0 + 1];
        c = wmma_f32(a, b, c);
    }
    const int n = col0 + mI, moff = (lane >> 4) * 8;
    const float bias = b_ih[n] + b_hh[n];
    const float wih  = W_ih[n];
#pragma unroll
    for (int r = 0; r < 8; ++r) {
        const int row = row0 + moff + r;
        gates[(long)row * GG + n] = c[r] + lstm_in[row] * wih + bias;
    }
}

// ---------------------------------------------------------------------------
// LSTM cell elementwise: i,f,g,o -> s' = sig(f)*s + sig(i)*tanh(g);
// d' = sig(o)*tanh(s')     (in-place update of d,s)
// ---------------------------------------------------------------------------
__global__ void lstm_elem_kernel(const float* __restrict__ gates,
                                 float* __restrict__ dm, float* __restrict__ sm)
{
    const int idx = blockIdx.x * 256 + threadIdx.x;   // b*128 + j
    const int b = idx >> 7, j = idx & 127;
    const float* g = gates + (long)b * GG;
    const float gi = g[j], gf = g[PP + j], gg = g[2 * PP + j], go = g[3 * PP + j];
    const float si = 1.0f / (1.0f + __expf(-gi));
    const float sf = 1.0f / (1.0f + __expf(-gf));
    const float so = 1.0f / (1.0f + __expf(-go));
    const float sn = sf * sm[idx] + si * tanhf(gg);
    sm[idx] = sn;
    dm[idx] = so * tanhf(sn);
}

// ---------------------------------------------------------------------------
// out[b,t] = FCout2_w . elu([d',C] @ FCout1_wT + FCout1_b) + FCout2_b
// M = 2048 (16/block), N = 128 full (8 waves), K = 128+128; fused ELU + dot
// ---------------------------------------------------------------------------
__global__ __launch_bounds__(256) void out_kernel(
    const float* __restrict__ dm, const float* __restrict__ C,
    const float* __restrict__ FCout1_w, const float* __restrict__ FCout1_b,
    const float* __restrict__ FCout2_w, const float* __restrict__ FCout2_b,
    float* __restrict__ out, int t)
{
    const int lane = threadIdx.x & 31, wave = threadIdx.x >> 5;
    const int row0 = blockIdx.x * 16, col0 = wave * 16;
    const int mI = lane & 15, kH = (lane >> 4) * 2;

    const float* ad = dm + (long)(row0 + mI) * PP + kH;
    const float* ac = C  + (long)(row0 + mI) * MM + kH;
    const float* bw = FCout1_w + (long)(col0 + mI) * (PP + MM) + kH;

    v8f c = {};
    for (int k0 = 0; k0 < PP; k0 += 4) {
        v2f a; a.x = ad[k0]; a.y = ad[k0 + 1];
        v2f b; b.x = bw[k0]; b.y = bw[k0 + 1];
        c = wmma_f32(a, b, c);
    }
    for (int k0 = 0; k0 < MM; k0 += 4) {
        v2f a; a.x = ac[k0]; a.y = ac[k0 + 1];
        v2f b; b.x = bw[PP + k0]; b.y = bw[PP + k0 + 1];
        c = wmma_f32(a, b, c);
    }

    // Fused epilogue: h = elu(c + b1); partial dot with FCout2_w
    const int n = col0 + mI, moff = (lane >> 4) * 8;
    const float b1 = FCout1_b[n];
    const float w2 = FCout2_w[n];
    float part[8];
#pragma unroll
    for (int r = 0; r < 8; ++r) {
        float hv = c[r] + b1;
        hv = (hv > 0.0f) ? hv : expm1f(hv);    // ELU(alpha=1)
        part[r] = hv * w2;
    }
    // reduce across the 16 lanes of each half-wave (same rows, distinct cols)
    for (int off = 1; off < 16; off <<= 1) {
#pragma unroll
        for (int r = 0; r < 8; ++r) part[r] += __shfl_xor(part[r], off, 32);
    }
    __shared__ float racc[8][16];              // [wave][row]
    if (mI == 0) {
#pragma unroll
        for (int r = 0; r < 8; ++r) racc[wave][moff + r] = part[r];
    }
    __syncthreads();
    if (threadIdx.x < 16) {
        float s = 0.0f;
#pragma unroll
        for (int w = 0; w < 8; ++w) s += racc[w][threadIdx.x];
        out[(long)(row0 + threadIdx.x) * LL + t] = s + FCout2_b[0];
    }
}

// ---------------------------------------------------------------------------
// Host-side orchestration (all on `stream`, graph-capture safe)
// ---------------------------------------------------------------------------
extern "C" void kernel_launch(void* const* d_in, const int* in_sizes, int n_in,
                              void* d_out, int out_size, void* d_ws, size_t ws_size,
                              hipStream_t stream)
{
    const float* H        = (const float*)d_in[0];   // (2048,64,128)
    const float* y        = (const float*)d_in[1];   // (2048,64)
    const float* FC1_w    = (const float*)d_in[2];   // (128,384)
    const float* FC1_b    = (const float*)d_in[3];   // (128,)
    const float* FC2_w    = (const float*)d_in[4];   // (1,128)
    const float* FC2_b    = (const float*)d_in[5];   // (1,)
    const float* FCin_w   = (const float*)d_in[6];   // (1,129)
    const float* FCin_b   = (const float*)d_in[7];   // (1,)
    const float* W_ih     = (const float*)d_in[8];   // (512,1)
    const float* W_hh     = (const float*)d_in[9];   // (512,128)
    const float* b_ih     = (const float*)d_in[10];  // (512,)
    const float* b_hh     = (const float*)d_in[11];  // (512,)
    const float* FCout1_w = (const float*)d_in[12];  // (128,256)
    const float* FCout1_b = (const float*)d_in[13];  // (128,)
    const float* FCout2_w = (const float*)d_in[14];  // (1,128)
    const float* FCout2_b = (const float*)d_in[15];  // (1,)
    float* out = (float*)d_out;                      // (2048,64)

    // Workspace partition (floats): ~75.5 MB total
    float* ws      = (float*)d_ws;
    float* HW      = ws;                       // BT*MM   = 16777216
    float* dm      = HW + (long)BT * MM;       // B*P     = 262144
    float* sm      = dm + BB * PP;             // B*P
    float* vv      = sm + BB * PP;             // B*MM
    float* Cm      = vv + BB * MM;             // B*MM
    float* gates   = Cm + BB * MM;             // B*512   = 1048576
    float* lstm_in = gates + (long)BB * GG;    // B       = 2048

    // d0 = s0 = 0
    zero_kernel<<<(2 * BB * PP + 255) / 256, 256, 0, stream>>>(dm, 2 * BB * PP);

    // One-time: HW = H @ W_H^T + b1  (time-invariant part of FC1)
    hw_gemm_kernel<<<BT / 16, 256, 0, stream>>>(H, FC1_w, FC1_b, HW);

    for (int t = 0; t < LL; ++t) {
        proj_kernel<<<BB / 16, 256, 0, stream>>>(dm, sm, FC1_w, vv);
        attn_kernel<<<BB, 256, 0, stream>>>(H, HW, vv, FC2_w, FC2_b,
                                            FCin_w, FCin_b, y, Cm, lstm_in, t);
        gates_gemm_kernel<<<dim3(BB / 16, 4), 256, 0, stream>>>(
            dm, W_hh, W_ih, b_ih, b_hh, lstm_in, gates);
        lstm_elem_kernel<<<(BB * PP) / 256, 256, 0, stream>>>(gates, dm, sm);
        out_kernel<<<BB / 16, 256, 0, stream>>>(dm, Cm, FCout1_w, FCout1_b,
                                                FCout2_w, FCout2_b, out, t);
    }
}